// FusionAwareInterpCVPR_70119636075171
// MI455X (gfx1250) — compile-verified
//
#include <hip/hip_runtime.h>

typedef __attribute__((ext_vector_type(2))) float v2f;
typedef __attribute__((ext_vector_type(8))) float v8f;

#define BSZ 2
#define HH 128
#define WW 192
#define NP 4096
#define CC 128
#define HWP (HH * WW)          // 24576
#define KPAD1 132              // 131 padded to multiple of 4
#define NEG_SLOPE 0.1f

// ---- workspace layout (in floats) ----
#define OFF_SAMPLED 0                                   // [BSZ][NP][CC]
#define OFF_IDX     (OFF_SAMPLED + BSZ * NP * CC)       // int [BSZ][HWP]
#define OFF_FEAT    (OFF_IDX + BSZ * HWP)               // [BSZ][132][HWP]
#define OFF_X1      (OFF_FEAT + BSZ * KPAD1 * HWP)      // [BSZ][128][HWP]
#define OFF_WP1     (OFF_X1 + BSZ * CC * HWP)           // 8*33*64
#define OFF_WP2     (OFF_WP1 + 8 * 33 * 64)             // 8*32*64
#define OFF_WP3     (OFF_WP2 + 8 * 32 * 64)             // 8*32*64

// LDS pair-row stride (dwords). 160 % 64 == 32 -> conflict-free b64 access.
#define PSTRIDE 160

// ---------------------------------------------------------------------------
// Bilinear sample of feat_2d at the NP uv points. One block per point,
// 128 threads = channels. Writes point-major so stores are coalesced.
__global__ void k_bilinear(const float* __restrict__ uv,
                           const float* __restrict__ f2d,
                           float* __restrict__ sampled) {
    const int n = blockIdx.x, b = blockIdx.y, c = threadIdx.x;
    const float x = uv[(b * 2 + 0) * NP + n];
    const float y = uv[(b * 2 + 1) * NP + n];
    const float x0f = floorf(x), y0f = floorf(y);
    const float wx = x - x0f, wy = y - y0f;
    int x0 = (int)x0f; x0 = min(max(x0, 0), WW - 1);
    const int x1 = min(x0 + 1, WW - 1);
    int y0 = (int)y0f; y0 = min(max(y0, 0), HH - 1);
    const int y1 = min(y0 + 1, HH - 1);
    const float* fp = f2d + (size_t)(b * CC + c) * HWP;
    const float v00 = fp[y0 * WW + x0];
    const float v01 = fp[y0 * WW + x1];
    const float v10 = fp[y1 * WW + x0];
    const float v11 = fp[y1 * WW + x1];
    const float r = v00 * (1.f - wx) * (1.f - wy) + v01 * wx * (1.f - wy)
                  + v10 * (1.f - wx) * wy         + v11 * wx * wy;
    sampled[((size_t)b * NP + n) * CC + c] = r;
}

// ---------------------------------------------------------------------------
// Nearest-neighbor search: uv (2*4096 floats = 32 KB) cached in LDS,
// one thread per pixel scans all 4096 points. Strict '<' => first argmin.
__global__ void __launch_bounds__(256) k_nn(const float* __restrict__ uv,
                                            int* __restrict__ nn_idx) {
    __shared__ float sx[NP];
    __shared__ float sy[NP];
    const int b = blockIdx.y;
    const int t = threadIdx.x;
    for (int i = t; i < NP; i += 256) {
        sx[i] = uv[(b * 2 + 0) * NP + i];
        sy[i] = uv[(b * 2 + 1) * NP + i];
    }
    __syncthreads();
    const int p = blockIdx.x * 256 + t;
    const float px = (float)(p % WW), py = (float)(p / WW);
    float best = 3.4e38f;
    int bi = 0;
    for (int n = 0; n < NP; ++n) {
        const float dx = sx[n] - px;
        const float dy = sy[n] - py;
        const float d = dx * dx + dy * dy;
        if (d < best) { best = d; bi = n; }
    }
    nn_idx[b * HWP + p] = bi;
}

// ---------------------------------------------------------------------------
// Feature rows 0,1 (offset), 2 (correlation), 131 (zero pad).
__global__ void __launch_bounds__(256) k_feat_head(const float* __restrict__ uv,
                                                   const float* __restrict__ f2d,
                                                   const float* __restrict__ sampled,
                                                   const int* __restrict__ nn_idx,
                                                   float* __restrict__ feat) {
    const int b = blockIdx.y;
    const int p = blockIdx.x * 256 + threadIdx.x;
    const int idx = nn_idx[b * HWP + p];
    const float px = (float)(p % WW), py = (float)(p / WW);
    const float ox = uv[(b * 2 + 0) * NP + idx] - px;
    const float oy = uv[(b * 2 + 1) * NP + idx] - py;
    const float* sp = sampled + ((size_t)b * NP + idx) * CC;
    const float* fp = f2d + (size_t)b * CC * HWP + p;
    float acc = 0.f;
    for (int c = 0; c < CC; ++c) acc += sp[c] * fp[(size_t)c * HWP];
    float* fb = feat + (size_t)b * KPAD1 * HWP;
    fb[0 * HWP + p] = ox;
    fb[1 * HWP + p] = oy;
    fb[2 * HWP + p] = acc * (1.0f / CC);
    fb[131 * HWP + p] = 0.f;
}

// ---------------------------------------------------------------------------
// Feature rows 3..130: gather feat_3d[:, nn_idx]. Coalesced writes.
__global__ void __launch_bounds__(256) k_feat_3d(const float* __restrict__ f3d,
                                                 const int* __restrict__ nn_idx,
                                                 float* __restrict__ feat) {
    const int b = blockIdx.z, c = blockIdx.y;
    const int p = blockIdx.x * 256 + threadIdx.x;
    const int idx = nn_idx[b * HWP + p];
    feat[((size_t)b * KPAD1 + 3 + c) * HWP + p] = f3d[((size_t)b * CC + c) * NP + idx];
}

// ---------------------------------------------------------------------------
// Pack weights into the V_WMMA_F32_16X16X4_F32 A-fragment lane layout:
// wp[mt][j][lane*2 + q] = w[16*mt + (lane&15)][4*j + 2*(lane>>4) + q]
// (zero-padded beyond Kin). A-loads in the GEMM become one coalesced b64.
__global__ void k_pack(const float* __restrict__ w, float* __restrict__ wp,
                       int Kin, int Ksteps) {
    const int gid = blockIdx.x * 256 + threadIdx.x;
    const int total = 8 * Ksteps * 32;
    if (gid >= total) return;
    const int lane = gid & 31;
    const int j = (gid >> 5) % Ksteps;
    const int mt = gid / (32 * Ksteps);
    const int row = mt * 16 + (lane & 15);
    const int k = 4 * j + ((lane >> 4) << 1);
    const float v0 = (k < Kin) ? w[row * Kin + k] : 0.f;
    const float v1 = (k + 1 < Kin) ? w[row * Kin + k + 1] : 0.f;
    wp[((mt * Ksteps + j) << 6) + lane * 2 + 0] = v0;
    wp[((mt * Ksteps + j) << 6) + lane * 2 + 1] = v1;
}

// ---------------------------------------------------------------------------
// GEMM + bias + LeakyReLU via V_WMMA_F32_16X16X4_F32.
// Block = 256 threads = 8 wave32; wave w owns output rows [16w,16w+16).
// Block tile: 128 rows x 64 cols.
//
// X tile is staged in LDS in a K-PAIR-INTERLEAVED layout:
//   xs[pair * PSTRIDE + 2*col + q] = X[2*pair + q][col0 + col],  q in {0,1}
// so each WMMA B fragment (K rows k,k+1 for this lane half) is ONE contiguous
// ds_load_b64 straight into an even-aligned VGPR pair -> no shuffle movs.
// PSTRIDE=160 (== 32 mod 64): lane halves land on disjoint bank halves,
// reads and the cooperative b64 writes are both bank-conflict-free.
template <int KPAD>
__global__ void __launch_bounds__(256)
k_gemm_lrelu(const float* __restrict__ Wp, const float* __restrict__ bias,
             const float* __restrict__ X, float* __restrict__ Y) {
    constexpr int NPAIR = KPAD / 2;
    constexpr int KSTEPS = KPAD / 4;
    __shared__ float xs[NPAIR * PSTRIDE];

    const int b = blockIdx.y;
    const int col0 = blockIdx.x * 64;
    const float* Xb = X + (size_t)b * KPAD * HWP;
    const int t = threadIdx.x;

    // Cooperative interleaved tile load: one ds_store_b64 per (pair, col).
    for (int idx = t; idx < NPAIR * 64; idx += 256) {
        const int pr = idx >> 6, c = idx & 63;
        v2f pv;
        pv.x = Xb[(size_t)(2 * pr + 0) * HWP + col0 + c];
        pv.y = Xb[(size_t)(2 * pr + 1) * HWP + col0 + c];
        *(v2f*)(xs + pr * PSTRIDE + 2 * c) = pv;
    }
    __syncthreads();

    const int wave = t >> 5, lane = t & 31;
    const int lrow = lane & 15;
    const float* wpw = Wp + (size_t)wave * KSTEPS * 64 + lane * 2;
    // lanes 0-15 read pair 2j (K = 4j,4j+1); lanes 16-31 pair 2j+1 (K = 4j+2,4j+3)
    const float* xrow = xs + (lane >> 4) * PSTRIDE + 2 * lrow;

    v8f acc0 = {}, acc1 = {}, acc2 = {}, acc3 = {};
#pragma unroll 4
    for (int j = 0; j < KSTEPS; ++j) {
        const v2f a = *(const v2f*)(wpw + j * 64);          // coalesced b64
        const float* xp = xrow + 2 * j * PSTRIDE;
        const v2f b0 = *(const v2f*)(xp + 0);               // ds_load_b64
        const v2f b1 = *(const v2f*)(xp + 32);
        const v2f b2 = *(const v2f*)(xp + 64);
        const v2f b3 = *(const v2f*)(xp + 96);
        acc0 = __builtin_amdgcn_wmma_f32_16x16x4_f32(false, a, false, b0, (short)0, acc0, false, false);
        acc1 = __builtin_amdgcn_wmma_f32_16x16x4_f32(false, a, false, b1, (short)0, acc1, false, false);
        acc2 = __builtin_amdgcn_wmma_f32_16x16x4_f32(false, a, false, b2, (short)0, acc2, false, false);
        acc3 = __builtin_amdgcn_wmma_f32_16x16x4_f32(false, a, false, b3, (short)0, acc3, false, false);
    }

    // Epilogue: C/D layout => VGPR v holds row 16*wave + v + 8*(lane>=16), col = lane&15.
    const int rbase = wave * 16 + ((lane >> 4) << 3);
    float* Yb = Y + (size_t)b * CC * HWP + col0 + lrow;
    const v8f accs[4] = {acc0, acc1, acc2, acc3};
#pragma unroll
    for (int s = 0; s < 4; ++s) {
#pragma unroll
        for (int v = 0; v < 8; ++v) {
            float y = accs[s][v] + bias[rbase + v];
            y = (y >= 0.f) ? y : NEG_SLOPE * y;
            Yb[(size_t)(rbase + v) * HWP + s * 16] = y;
        }
    }
}

// ---------------------------------------------------------------------------
extern "C" void kernel_launch(void* const* d_in, const int* in_sizes, int n_in,
                              void* d_out, int out_size, void* d_ws, size_t ws_size,
                              hipStream_t stream) {
    const float* uv  = (const float*)d_in[0];
    const float* f2d = (const float*)d_in[1];
    const float* f3d = (const float*)d_in[2];
    const float* w1  = (const float*)d_in[3];
    const float* b1  = (const float*)d_in[4];
    const float* w2  = (const float*)d_in[5];
    const float* b2  = (const float*)d_in[6];
    const float* w3  = (const float*)d_in[7];
    const float* b3  = (const float*)d_in[8];
    float* out = (float*)d_out;
    float* ws  = (float*)d_ws;

    float* sampled = ws + OFF_SAMPLED;
    int*   nnidx   = (int*)(ws + OFF_IDX);
    float* feat    = ws + OFF_FEAT;   // [BSZ][132][HWP]; also reused as x2
    float* x1      = ws + OFF_X1;     // [BSZ][128][HWP]
    float* wp1     = ws + OFF_WP1;
    float* wp2     = ws + OFF_WP2;
    float* wp3     = ws + OFF_WP3;

    // Weight packing (tiny)
    k_pack<<<(8 * 33 * 32 + 255) / 256, 256, 0, stream>>>(w1, wp1, 131, 33);
    k_pack<<<(8 * 32 * 32 + 255) / 256, 256, 0, stream>>>(w2, wp2, 128, 32);
    k_pack<<<(8 * 32 * 32 + 255) / 256, 256, 0, stream>>>(w3, wp3, 128, 32);

    // Stage 1: sampling + NN search
    k_bilinear<<<dim3(NP, BSZ), CC, 0, stream>>>(uv, f2d, sampled);
    k_nn<<<dim3(HWP / 256, BSZ), 256, 0, stream>>>(uv, nnidx);

    // Stage 2: assemble 132 x P feature matrix (row 131 zero pad)
    k_feat_head<<<dim3(HWP / 256, BSZ), 256, 0, stream>>>(uv, f2d, sampled, nnidx, feat);
    k_feat_3d<<<dim3(HWP / 256, CC, BSZ), 256, 0, stream>>>(f3d, nnidx, feat);

    // Stage 3: three WMMA GEMM + LeakyReLU layers.
    // Layer 2 writes into the (now dead) feat buffer region as [b][128][P].
    k_gemm_lrelu<KPAD1><<<dim3(HWP / 64, BSZ), 256, 0, stream>>>(wp1, b1, feat, x1);
    k_gemm_lrelu<CC>   <<<dim3(HWP / 64, BSZ), 256, 0, stream>>>(wp2, b2, x1, feat);
    k_gemm_lrelu<CC>   <<<dim3(HWP / 64, BSZ), 256, 0, stream>>>(wp3, b3, feat, out);
}